// HierarchicalAttention_18966575579198
// MI455X (gfx1250) — compile-verified
//
#include <hip/hip_runtime.h>
#include <hip/hip_bf16.h>

typedef float v2f __attribute__((ext_vector_type(2)));
typedef float v8f __attribute__((ext_vector_type(8)));

#define S_LEN 2048
#define E_DIM 1024
#define NH    16
#define HD    64
#define NBATCH 2

__device__ __forceinline__ v8f wmma_f32(v2f a, v2f b, v8f c) {
    // V_WMMA_F32_16X16X4_F32 : D = A(16x4) * B(4x16) + C(16x16), full fp32
    return __builtin_amdgcn_wmma_f32_16x16x4_f32(false, a, false, b, (short)0, c, false, false);
}

// ---------------------------------------------------------------------------
// Kernel 0: precompute RoPE table, interleaved [S, 32, {cos,sin}] (512 KB).
// Uses precise libm sincosf once; spills here are irrelevant. The hot GEMM
// kernel then does pure b64 loads + FMAs (no calls, no VGPR spills).
// ---------------------------------------------------------------------------
__global__ void rope_table_kernel(float* __restrict__ tab)
{
    const int tid = blockIdx.x * blockDim.x + threadIdx.x;   // 0 .. S*32-1
    const int j = tid & 31;
    const int s = tid >> 5;
    // inv_freq[j] = 10000^(-j/32) = 10^(-j/8)
    const float inv = exp10f(-(float)j / 8.0f);
    float sn, cs;
    sincosf((float)s * inv, &sn, &cs);
    tab[(size_t)tid * 2 + 0] = cs;
    tab[(size_t)tid * 2 + 1] = sn;
}

// ---------------------------------------------------------------------------
// Kernel 1: fused QKV projection (y = x @ W^T) + RoPE on q,k.
// grid = (48, 128): x = col-tile (16 per matrix * 3 matrices), y = 32-row tile.
// One wave (32 threads) computes a 32x64 output tile with fp32 WMMA.
// Each 64-col tile == one head. Output layout [B,H,S,D].
// ---------------------------------------------------------------------------
__global__ void qkv_rope_kernel(const float* __restrict__ x,
                                const float* __restrict__ wq,
                                const float* __restrict__ wk,
                                const float* __restrict__ wv,
                                const float* __restrict__ rope_tab,
                                float* __restrict__ q_ws,
                                float* __restrict__ k_ws,
                                float* __restrict__ v_ws)
{
    const int lane = threadIdx.x;
    const int half = lane >> 4;     // 0: K=0,1 / 1: K=2,3 of each WMMA step
    const int l16  = lane & 15;

    const int colTile = blockIdx.x;        // 0..47
    const int mat     = colTile >> 4;      // 0=q,1=k,2=v
    const int head    = colTile & 15;      // each 64-col tile is one head
    const int colBase = head * HD;
    const int rowBase = blockIdx.y * 32;   // rows in [0, B*S)

    const float* w    = (mat == 0) ? wq : (mat == 1) ? wk : wv;
    float*       outp = (mat == 0) ? q_ws : (mat == 1) ? k_ws : v_ws;

    v8f acc[2][4];
#pragma unroll
    for (int mt = 0; mt < 2; ++mt)
#pragma unroll
        for (int nt = 0; nt < 4; ++nt) acc[mt][nt] = (v8f)0.0f;

    const float* xr0 = x + (size_t)(rowBase + l16) * E_DIM + 2 * half;
    const float* xr1 = xr0 + (size_t)16 * E_DIM;
    const float* wr0 = w + (size_t)(colBase + 0 * 16 + l16) * E_DIM + 2 * half;
    const float* wr1 = w + (size_t)(colBase + 1 * 16 + l16) * E_DIM + 2 * half;
    const float* wr2 = w + (size_t)(colBase + 2 * 16 + l16) * E_DIM + 2 * half;
    const float* wr3 = w + (size_t)(colBase + 3 * 16 + l16) * E_DIM + 2 * half;

    for (int k0 = 0; k0 < E_DIM; k0 += 4) {
        v2f a0 = *(const v2f*)(xr0 + k0);
        v2f a1 = *(const v2f*)(xr1 + k0);
        v2f b0 = *(const v2f*)(wr0 + k0);
        v2f b1 = *(const v2f*)(wr1 + k0);
        v2f b2 = *(const v2f*)(wr2 + k0);
        v2f b3 = *(const v2f*)(wr3 + k0);
        acc[0][0] = wmma_f32(a0, b0, acc[0][0]);
        acc[1][0] = wmma_f32(a1, b0, acc[1][0]);
        acc[0][1] = wmma_f32(a0, b1, acc[0][1]);
        acc[1][1] = wmma_f32(a1, b1, acc[1][1]);
        acc[0][2] = wmma_f32(a0, b2, acc[0][2]);
        acc[1][2] = wmma_f32(a1, b2, acc[1][2]);
        acc[0][3] = wmma_f32(a0, b3, acc[0][3]);
        acc[1][3] = wmma_f32(a1, b3, acc[1][3]);
    }

    const int b     = rowBase >> 11;        // batch (S=2048)
    const int sBase = rowBase & (S_LEN - 1);

    if (mat < 2) {
        // RoPE: d = nt*16 + l16; pairs (d, d+32) are (nt, nt+2), same lane/VGPR.
        // cos/sin from precomputed table: tab[(s*32 + j)*2 + {0,1}], j = d & 31.
#pragma unroll
        for (int mt = 0; mt < 2; ++mt) {
#pragma unroll
            for (int r = 0; r < 8; ++r) {
                const int s = sBase + mt * 16 + r + 8 * half;
                v2f cs0 = *(const v2f*)(rope_tab + ((size_t)s * 32 + l16) * 2);
                v2f cs1 = *(const v2f*)(rope_tab + ((size_t)s * 32 + 16 + l16) * 2);
                const float c0 = cs0.x, s0 = cs0.y;
                const float c1 = cs1.x, s1 = cs1.y;
                float a0 = acc[mt][0][r], a1 = acc[mt][1][r];
                float a2 = acc[mt][2][r], a3 = acc[mt][3][r];
                acc[mt][0][r] = a0 * c0 - a2 * s0;
                acc[mt][1][r] = a1 * c1 - a3 * s1;
                acc[mt][2][r] = a2 * c0 + a0 * s0;
                acc[mt][3][r] = a3 * c1 + a1 * s1;
            }
        }
    }

    float* hb = outp + ((size_t)(b * NH + head) * S_LEN) * HD;
#pragma unroll
    for (int mt = 0; mt < 2; ++mt)
#pragma unroll
        for (int r = 0; r < 8; ++r) {
            const int s = sBase + mt * 16 + r + 8 * half;
#pragma unroll
            for (int nt = 0; nt < 4; ++nt)
                hb[(size_t)s * HD + nt * 16 + l16] = acc[mt][nt][r];
        }
}

// ---------------------------------------------------------------------------
// Kernel 2: flash attention per (b,h). grid = (128, 32): x = 16-row q tile,
// y = b*16+h. One wave streams 64-key tiles with online softmax; scores never
// touch memory. P is staged through a padded LDS tile to re-layout C->A.
// Output ctx in [B,S,E] layout.
// ---------------------------------------------------------------------------
__global__ void flash_attn_kernel(const float* __restrict__ q_ws,
                                  const float* __restrict__ k_ws,
                                  const float* __restrict__ v_ws,
                                  float* __restrict__ ctx_ws)
{
    __shared__ float p_lds[16 * 68];   // 68-float row stride: conflict-free b64 reads

    const int lane = threadIdx.x;
    const int half = lane >> 4;
    const int l16  = lane & 15;

    const int bh    = blockIdx.y;           // 0..31
    const int b     = bh >> 4;
    const int head  = bh & 15;
    const int qBase = blockIdx.x * 16;
    const size_t base = (size_t)bh * S_LEN * HD;

    // Q fragments in A-layout for all 16 K-steps, pre-scaled by 1/sqrt(D)
    v2f qa[16];
    {
        const float* qp = q_ws + base + (size_t)(qBase + l16) * HD + 2 * half;
#pragma unroll
        for (int ds = 0; ds < 16; ++ds) {
            v2f t = *(const v2f*)(qp + 4 * ds);
            qa[ds] = t * 0.125f;
        }
    }

    v8f ctx[4];
#pragma unroll
    for (int nt = 0; nt < 4; ++nt) ctx[nt] = (v8f)0.0f;
    float mrow[8], lrow[8];
#pragma unroll
    for (int r = 0; r < 8; ++r) { mrow[r] = -3.0e38f; lrow[r] = 0.0f; }

    for (int kb = 0; kb < S_LEN; kb += 64) {
        // ---- scores = Qs @ K^T (16 x 64keys) ----
        v8f sc[4];
#pragma unroll
        for (int nt = 0; nt < 4; ++nt) sc[nt] = (v8f)0.0f;
        const float* kp = k_ws + base + (size_t)(kb + l16) * HD + 2 * half;
#pragma unroll
        for (int ds = 0; ds < 16; ++ds) {
#pragma unroll
            for (int nt = 0; nt < 4; ++nt) {
                v2f bf = *(const v2f*)(kp + (size_t)(nt * 16) * HD + 4 * ds);
                sc[nt] = wmma_f32(qa[ds], bf, sc[nt]);
            }
        }

        // ---- online softmax: row max over 4 tiles + 16 lanes of this half ----
        float tval[8];
#pragma unroll
        for (int r = 0; r < 8; ++r)
            tval[r] = fmaxf(fmaxf(sc[0][r], sc[1][r]), fmaxf(sc[2][r], sc[3][r]));
#pragma unroll
        for (int off = 1; off < 16; off <<= 1)
#pragma unroll
            for (int r = 0; r < 8; ++r)
                tval[r] = fmaxf(tval[r], __shfl_xor(tval[r], off, 32));

        float scale[8];
#pragma unroll
        for (int r = 0; r < 8; ++r) {
            const float mnew = fmaxf(mrow[r], tval[r]);
            scale[r] = __expf(mrow[r] - mnew);
            mrow[r]  = mnew;
        }
#pragma unroll
        for (int nt = 0; nt < 4; ++nt)
#pragma unroll
            for (int r = 0; r < 8; ++r)
                sc[nt][r] = __expf(sc[nt][r] - mrow[r]);

        float psum[8];
#pragma unroll
        for (int r = 0; r < 8; ++r)
            psum[r] = (sc[0][r] + sc[1][r]) + (sc[2][r] + sc[3][r]);
#pragma unroll
        for (int off = 1; off < 16; off <<= 1)
#pragma unroll
            for (int r = 0; r < 8; ++r)
                psum[r] += __shfl_xor(psum[r], off, 32);
#pragma unroll
        for (int r = 0; r < 8; ++r)
            lrow[r] = lrow[r] * scale[r] + psum[r];

        // rescale running context
#pragma unroll
        for (int nt = 0; nt < 4; ++nt)
#pragma unroll
            for (int r = 0; r < 8; ++r)
                ctx[nt][r] *= scale[r];

        // ---- stage P (C-layout) into LDS, reread in A-layout ----
        __syncthreads();
#pragma unroll
        for (int nt = 0; nt < 4; ++nt)
#pragma unroll
            for (int r = 0; r < 8; ++r)
                p_lds[(r + 8 * half) * 68 + nt * 16 + l16] = sc[nt][r];
        __syncthreads();

        // ---- ctx += P @ V ----
#pragma unroll
        for (int ks = 0; ks < 16; ++ks) {
            v2f pa = *(const v2f*)(&p_lds[l16 * 68 + 4 * ks + 2 * half]);
            const float* vr = v_ws + base + (size_t)(kb + 4 * ks + 2 * half) * HD;
#pragma unroll
            for (int nt = 0; nt < 4; ++nt) {
                v2f bf;
                bf.x = vr[nt * 16 + l16];
                bf.y = vr[HD + nt * 16 + l16];
                ctx[nt] = wmma_f32(pa, bf, ctx[nt]);
            }
        }
    }

    // ---- normalize and store ctx -> [B,S,E] ----
    float inv[8];
#pragma unroll
    for (int r = 0; r < 8; ++r) inv[r] = 1.0f / lrow[r];
    float* op = ctx_ws + ((size_t)b * S_LEN) * E_DIM + head * HD;
#pragma unroll
    for (int r = 0; r < 8; ++r) {
        const int s = qBase + r + 8 * half;
#pragma unroll
        for (int nt = 0; nt < 4; ++nt)
            op[(size_t)s * E_DIM + nt * 16 + l16] = ctx[nt][r] * inv[r];
    }
}

// ---------------------------------------------------------------------------
// Kernel 3: output projection out = ctx @ wo^T. grid = (16, 128).
// One wave computes a 32x64 tile with fp32 WMMA.
// ---------------------------------------------------------------------------
__global__ void out_proj_kernel(const float* __restrict__ ctx_ws,
                                const float* __restrict__ wo,
                                float* __restrict__ out)
{
    const int lane = threadIdx.x;
    const int half = lane >> 4;
    const int l16  = lane & 15;

    const int colBase = blockIdx.x * 64;
    const int rowBase = blockIdx.y * 32;

    v8f acc[2][4];
#pragma unroll
    for (int mt = 0; mt < 2; ++mt)
#pragma unroll
        for (int nt = 0; nt < 4; ++nt) acc[mt][nt] = (v8f)0.0f;

    const float* xr0 = ctx_ws + (size_t)(rowBase + l16) * E_DIM + 2 * half;
    const float* xr1 = xr0 + (size_t)16 * E_DIM;
    const float* wr0 = wo + (size_t)(colBase + 0 * 16 + l16) * E_DIM + 2 * half;
    const float* wr1 = wo + (size_t)(colBase + 1 * 16 + l16) * E_DIM + 2 * half;
    const float* wr2 = wo + (size_t)(colBase + 2 * 16 + l16) * E_DIM + 2 * half;
    const float* wr3 = wo + (size_t)(colBase + 3 * 16 + l16) * E_DIM + 2 * half;

    for (int k0 = 0; k0 < E_DIM; k0 += 4) {
        v2f a0 = *(const v2f*)(xr0 + k0);
        v2f a1 = *(const v2f*)(xr1 + k0);
        v2f b0 = *(const v2f*)(wr0 + k0);
        v2f b1 = *(const v2f*)(wr1 + k0);
        v2f b2 = *(const v2f*)(wr2 + k0);
        v2f b3 = *(const v2f*)(wr3 + k0);
        acc[0][0] = wmma_f32(a0, b0, acc[0][0]);
        acc[1][0] = wmma_f32(a1, b0, acc[1][0]);
        acc[0][1] = wmma_f32(a0, b1, acc[0][1]);
        acc[1][1] = wmma_f32(a1, b1, acc[1][1]);
        acc[0][2] = wmma_f32(a0, b2, acc[0][2]);
        acc[1][2] = wmma_f32(a1, b2, acc[1][2]);
        acc[0][3] = wmma_f32(a0, b3, acc[0][3]);
        acc[1][3] = wmma_f32(a1, b3, acc[1][3]);
    }

#pragma unroll
    for (int mt = 0; mt < 2; ++mt)
#pragma unroll
        for (int r = 0; r < 8; ++r) {
            const int row = rowBase + mt * 16 + r + 8 * half;
#pragma unroll
            for (int nt = 0; nt < 4; ++nt)
                out[(size_t)row * E_DIM + colBase + nt * 16 + l16] = acc[mt][nt][r];
        }
}

// ---------------------------------------------------------------------------
extern "C" void kernel_launch(void* const* d_in, const int* in_sizes, int n_in,
                              void* d_out, int out_size, void* d_ws, size_t ws_size,
                              hipStream_t stream) {
    const float* x  = (const float*)d_in[0];
    const float* wq = (const float*)d_in[1];
    const float* wk = (const float*)d_in[2];
    const float* wv = (const float*)d_in[3];
    const float* wo = (const float*)d_in[4];
    float* out = (float*)d_out;

    const size_t QKV_ELEMS = (size_t)NBATCH * NH * S_LEN * HD;  // 4,194,304
    float* ws    = (float*)d_ws;
    float* q_ws  = ws;
    float* k_ws  = ws + QKV_ELEMS;
    float* v_ws  = ws + 2 * QKV_ELEMS;
    float* c_ws  = ws + 3 * QKV_ELEMS;
    float* tab   = ws + 4 * QKV_ELEMS;      // S*32*2 floats = 512 KB

    rope_table_kernel<<<dim3((S_LEN * 32) / 256), dim3(256), 0, stream>>>(tab);
    qkv_rope_kernel<<<dim3(48, 128), dim3(32), 0, stream>>>(x, wq, wk, wv, tab, q_ws, k_ws, v_ws);
    flash_attn_kernel<<<dim3(128, 32), dim3(32), 0, stream>>>(q_ws, k_ws, v_ws, c_ws);
    out_proj_kernel<<<dim3(16, 128), dim3(32), 0, stream>>>(c_ws, wo, out);
}